// GIN_18305150616170
// MI455X (gfx1250) — compile-verified
//
#include <hip/hip_runtime.h>
#include <hip/hip_bf16.h>

typedef __attribute__((ext_vector_type(2))) float v2f;
typedef __attribute__((ext_vector_type(8))) float v8f;

#define N_NODES   50000
#define N_EDGES   400000
#define N_GRAPHS  256
#define IN_DIM    128
#define HID       256
#define OUT_DIM   64
#define BN_EPS    1e-5f

// ---------------------------------------------------------------------------
// Simple elementwise helpers
// ---------------------------------------------------------------------------
__global__ void zero_f32(float* __restrict__ p, int n) {
  int i = blockIdx.x * blockDim.x + threadIdx.x;
  if (i < n) p[i] = 0.0f;
}

__global__ void copy_f32v4(float* __restrict__ dst, const float* __restrict__ src, int n4) {
  int i = blockIdx.x * blockDim.x + threadIdx.x;
  if (i < n4) ((float4*)dst)[i] = ((const float4*)src)[i];
}

// ---------------------------------------------------------------------------
// Edge scatter: Y[dst] += X[src]  (one thread per edge x float4-chunk)
// cshift = log2(dim/4)
// ---------------------------------------------------------------------------
__global__ void edge_scatter(float* __restrict__ Y, const float* __restrict__ X,
                             const int* __restrict__ src, const int* __restrict__ dst,
                             int nE, int cshift) {
  int tid = blockIdx.x * blockDim.x + threadIdx.x;
  if (tid >= (nE << cshift)) return;
  int e = tid >> cshift;
  int c = tid & ((1 << cshift) - 1);
  int s = src[e], d = dst[e];
  int dim = 4 << cshift;
  const float4 v = ((const float4*)(X + (size_t)s * dim))[c];
  float* p = Y + (size_t)d * dim + c * 4;
  atomicAdd(p + 0, v.x);
  atomicAdd(p + 1, v.y);
  atomicAdd(p + 2, v.z);
  atomicAdd(p + 3, v.w);
}

// ---------------------------------------------------------------------------
// Graph sum-pooling: P[gid[v]] += X[v]
// ---------------------------------------------------------------------------
__global__ void pool_rows(float* __restrict__ P, const float* __restrict__ X,
                          const int* __restrict__ gid, int M, int cshift) {
  int tid = blockIdx.x * blockDim.x + threadIdx.x;
  if (tid >= (M << cshift)) return;
  int v = tid >> cshift;
  int c = tid & ((1 << cshift) - 1);
  int g = gid[v];
  int dim = 4 << cshift;
  const float4 val = ((const float4*)(X + (size_t)v * dim))[c];
  float* p = P + (size_t)g * dim + c * 4;
  atomicAdd(p + 0, val.x);
  atomicAdd(p + 1, val.y);
  atomicAdd(p + 2, val.z);
  atomicAdd(p + 3, val.w);
}

// ---------------------------------------------------------------------------
// Repack row-major W (K x N) into WMMA f32 16x16x4 B-fragment order:
//   frag f = (kk_step * (N/16) + ntile); element (lane, j):
//     Wp[(f*32 + lane)*2 + j] = W[(kk_step*4 + (lane>>4)*2 + j) * N + ntile*16 + (lane&15)]
// ---------------------------------------------------------------------------
__global__ void repack_w(const float* __restrict__ W, float* __restrict__ Wp, int K, int N) {
  int tid = blockIdx.x * blockDim.x + threadIdx.x;
  if (tid >= K * N) return;
  int f = tid >> 6;
  int r = tid & 63;
  int L = r >> 1, j = r & 1;
  int ntiles = N >> 4;
  int kk = f / ntiles, nt = f - kk * ntiles;
  int k = kk * 4 + ((L >> 4) << 1) + j;
  int n = nt * 16 + (L & 15);
  Wp[tid] = W[(size_t)k * N + n];
}

// ---------------------------------------------------------------------------
// Main GEMM: C[M x 256] = X[M x K] @ W[K x 256]   (W pre-packed)
// 8 waves/block in a 2(M) x 4(N) grid; wave tile 32 x 64 (2x4 WMMA accs).
// Block tile: 64 rows x 256 cols -> 1-D grid over M.
// Out-of-range rows: addresses clamped to row M-1; their results land only in
// D rows >= M (WMMA row-locality), which are never stored -> no zeroing needed.
// ---------------------------------------------------------------------------
__global__ __launch_bounds__(256) void gemm_wmma_f32(
    const float* __restrict__ X, const float* __restrict__ Wp,
    float* __restrict__ C, int M, int K) {
  const int N = HID;
  const int lane  = threadIdx.x & 31;
  const int wave  = threadIdx.x >> 5;
  const int wm    = wave & 1;
  const int wn    = wave >> 1;            // 0..3
  const int m_base = blockIdx.x * 64 + wm * 32;
  const int n_base = wn * 64;
  const int lrow  = lane & 15;
  const int khalf = (lane >> 4) << 1;     // 0 or 2
  const int ntiles = N >> 4;              // 16
  const int nt0 = n_base >> 4;

  v8f acc[2][4] = {};

  const int r0 = m_base + lrow;
  const int r1 = m_base + 16 + lrow;
  const int r0c = (r0 < M) ? r0 : (M - 1);
  const int r1c = (r1 < M) ? r1 : (M - 1);
  const float* xr0 = X + (size_t)r0c * K + khalf;
  const float* xr1 = X + (size_t)r1c * K + khalf;

  for (int kk = 0; kk < K; kk += 4) {
    v2f a0 = *(const v2f*)(xr0 + kk);
    v2f a1 = *(const v2f*)(xr1 + kk);
    const float* bp = Wp + (((size_t)(kk >> 2) * ntiles + nt0) << 6) + (lane << 1);
    v2f b0 = *(const v2f*)(bp);
    v2f b1 = *(const v2f*)(bp + 64);
    v2f b2 = *(const v2f*)(bp + 128);
    v2f b3 = *(const v2f*)(bp + 192);
    acc[0][0] = __builtin_amdgcn_wmma_f32_16x16x4_f32(false, a0, false, b0, (short)0, acc[0][0], false, false);
    acc[0][1] = __builtin_amdgcn_wmma_f32_16x16x4_f32(false, a0, false, b1, (short)0, acc[0][1], false, false);
    acc[0][2] = __builtin_amdgcn_wmma_f32_16x16x4_f32(false, a0, false, b2, (short)0, acc[0][2], false, false);
    acc[0][3] = __builtin_amdgcn_wmma_f32_16x16x4_f32(false, a0, false, b3, (short)0, acc[0][3], false, false);
    acc[1][0] = __builtin_amdgcn_wmma_f32_16x16x4_f32(false, a1, false, b0, (short)0, acc[1][0], false, false);
    acc[1][1] = __builtin_amdgcn_wmma_f32_16x16x4_f32(false, a1, false, b1, (short)0, acc[1][1], false, false);
    acc[1][2] = __builtin_amdgcn_wmma_f32_16x16x4_f32(false, a1, false, b2, (short)0, acc[1][2], false, false);
    acc[1][3] = __builtin_amdgcn_wmma_f32_16x16x4_f32(false, a1, false, b3, (short)0, acc[1][3], false, false);
  }

  const int rhalf = (lane >> 4) << 3;     // 0 or 8
  #pragma unroll
  for (int mi = 0; mi < 2; mi++) {
    #pragma unroll
    for (int i = 0; i < 8; i++) {
      int row = m_base + mi * 16 + rhalf + i;
      if (row < M) {
        float* cp = C + (size_t)row * N + n_base + lrow;
        cp[0]  = acc[mi][0][i];
        cp[16] = acc[mi][1][i];
        cp[32] = acc[mi][2][i];
        cp[48] = acc[mi][3][i];
      }
    }
  }
}

// ---------------------------------------------------------------------------
// Column statistics over M rows of a (M x 256) matrix, f64 atomic accumulate.
// ---------------------------------------------------------------------------
__global__ __launch_bounds__(256) void col_stats(const float* __restrict__ X,
                                                 double* __restrict__ st, int M) {
  int c = threadIdx.x;
  double s = 0.0, s2 = 0.0;
  for (int r = blockIdx.x; r < M; r += gridDim.x) {
    float v = X[(size_t)r * HID + c];
    s += (double)v;
    s2 += (double)v * (double)v;
  }
  atomicAdd(&st[c], s);
  atomicAdd(&st[HID + c], s2);
}

// Fold BN into per-column scale/shift; self-reset stats for next use.
__global__ void bn_finalize(double* __restrict__ st, const float* __restrict__ g,
                            const float* __restrict__ b, float* __restrict__ coeff, int M) {
  int c = threadIdx.x;
  double m = st[c] / (double)M;
  double v = st[HID + c] / (double)M - m * m;
  float a = g[c] * rsqrtf((float)v + BN_EPS);
  coeff[c] = a;
  coeff[HID + c] = b[c] - (float)m * a;
  st[c] = 0.0;
  st[HID + c] = 0.0;
}

// y = relu(a[col] * x + b[col]), in place, per float4 chunk (256 cols).
__global__ void bn_apply_relu(float* __restrict__ X, const float* __restrict__ coeff, int M) {
  int tid = blockIdx.x * blockDim.x + threadIdx.x;
  if (tid >= M * (HID / 4)) return;
  int c4 = (tid & 63) << 2;
  float4 v = ((float4*)X)[tid];
  v.x = fmaxf(coeff[c4 + 0] * v.x + coeff[HID + c4 + 0], 0.0f);
  v.y = fmaxf(coeff[c4 + 1] * v.y + coeff[HID + c4 + 1], 0.0f);
  v.z = fmaxf(coeff[c4 + 2] * v.z + coeff[HID + c4 + 2], 0.0f);
  v.w = fmaxf(coeff[c4 + 3] * v.w + coeff[HID + c4 + 3], 0.0f);
  ((float4*)X)[tid] = v;
}

// ---------------------------------------------------------------------------
// Prediction GEMM: S[256 x 64] (+)= P[256 x K] @ W[K x 64] + bias
// One wave per 16x16 tile; 64 tiles; 4 waves/block.
// ---------------------------------------------------------------------------
__global__ __launch_bounds__(128) void pred_gemm(const float* __restrict__ P,
                                                 const float* __restrict__ Wp,
                                                 const float* __restrict__ bias,
                                                 float* __restrict__ S,
                                                 int K, int accumulate) {
  const int lane = threadIdx.x & 31;
  const int wave = threadIdx.x >> 5;
  const int tile = blockIdx.x * 4 + wave;      // 0..63
  const int mt = tile >> 2;
  const int nt = tile & 3;
  const int m_base = mt << 4;
  const int n_base = nt << 4;
  const int lrow  = lane & 15;
  const int khalf = (lane >> 4) << 1;

  v8f acc = {};
  const float* pr = P + (size_t)(m_base + lrow) * K + khalf;
  for (int kk = 0; kk < K; kk += 4) {
    v2f a = *(const v2f*)(pr + kk);
    v2f b = *(const v2f*)(Wp + ((((size_t)(kk >> 2) << 2) + nt) << 6) + (lane << 1));
    acc = __builtin_amdgcn_wmma_f32_16x16x4_f32(false, a, false, b, (short)0, acc, false, false);
  }

  const int col = n_base + lrow;
  const float bb = bias[col];
  const int rhalf = (lane >> 4) << 3;
  #pragma unroll
  for (int i = 0; i < 8; i++) {
    size_t idx = (size_t)(m_base + rhalf + i) * OUT_DIM + col;
    float prev = accumulate ? S[idx] : 0.0f;
    S[idx] = prev + acc[i] + bb;
  }
}

// ---------------------------------------------------------------------------
// Host orchestration
// ---------------------------------------------------------------------------
static inline int ceil_div(int a, int b) { return (a + b - 1) / b; }

extern "C" void kernel_launch(void* const* d_in, const int* in_sizes, int n_in,
                              void* d_out, int out_size, void* d_ws, size_t ws_size,
                              hipStream_t stream) {
  (void)in_sizes; (void)n_in; (void)out_size; (void)ws_size;

  // ---- input map (setup_inputs flatten order) ----
  const float* h = (const float*)d_in[0];
  // gin layer l: base = 1 + 22*l
  //   +0..4  w[0..4]      +5..9  b[0..4] (all zero; cancel under BN -> unused)
  //   +10..13 mlp_bn_g    +14..17 mlp_bn_b
  //   +18 apply_bn_g  +19 apply_bn_b  +20 out_bn_g  +21 out_bn_b
  const int PRED_W = 1 + 22 * 3;          // 67
  const int PRED_B = PRED_W + 4;          // 71
  const int* esrc = (const int*)d_in[75];
  const int* edst = (const int*)d_in[76];
  const int* gids = (const int*)d_in[77];
  float* score = (float*)d_out;

  // ---- workspace carve ----
  size_t off = 0;
  auto carve = [&](size_t bytes) -> char* {
    char* p = (char*)d_ws + off;
    off += (bytes + 255) & ~(size_t)255;
    return p;
  };
  float*  bufA   = (float*)carve((size_t)N_NODES * HID * sizeof(float));
  float*  bufB   = (float*)carve((size_t)N_NODES * HID * sizeof(float));
  float*  wpack  = (float*)carve((size_t)1007616 * sizeof(float));
  double* stats  = (double*)carve(2 * HID * sizeof(double));
  float*  coeff  = (float*)carve(2 * HID * sizeof(float));
  float*  pooled = (float*)carve((size_t)N_GRAPHS * HID * sizeof(float));

  // ---- zero BN stats (finalize self-resets afterwards) ----
  zero_f32<<<ceil_div(2 * HID * 2, 256), 256, 0, stream>>>((float*)stats, 2 * HID * 2);

  // ---- repack all weights into WMMA B-fragment order ----
  size_t woff_gin[3][5];
  size_t woff_pred[4];
  {
    size_t o = 0;
    for (int l = 0; l < 3; l++)
      for (int i = 0; i < 5; i++) {
        int K = (l == 0 && i == 0) ? IN_DIM : HID;
        woff_gin[l][i] = o;
        int total = K * HID;
        repack_w<<<ceil_div(total, 256), 256, 0, stream>>>(
            (const float*)d_in[1 + 22 * l + i], wpack + o, K, HID);
        o += (size_t)total;
      }
    for (int i = 0; i < 4; i++) {
      int K = (i == 0) ? IN_DIM : HID;
      woff_pred[i] = o;
      int total = K * OUT_DIM;
      repack_w<<<ceil_div(total, 256), 256, 0, stream>>>(
          (const float*)d_in[PRED_W + i], wpack + o, K, OUT_DIM);
      o += (size_t)total;
    }
  }

  // BN pass helper: stats -> finalize -> fused scale/shift+relu in place
  auto bn_pass = [&](float* X, const float* g, const float* b) {
    col_stats<<<256, 256, 0, stream>>>(X, stats, N_NODES);
    bn_finalize<<<1, HID, 0, stream>>>(stats, g, b, coeff, N_NODES);
    bn_apply_relu<<<ceil_div(N_NODES * (HID / 4), 256), 256, 0, stream>>>(X, coeff, N_NODES);
  };
  // pooling + prediction for one hidden state
  auto pool_pred = [&](const float* X, int din, int layer, int accumulate) {
    zero_f32<<<ceil_div(N_GRAPHS * din, 256), 256, 0, stream>>>(pooled, N_GRAPHS * din);
    int cshift = (din == IN_DIM) ? 5 : 6;     // din/4 = 32 or 64
    pool_rows<<<ceil_div(N_NODES << cshift, 256), 256, 0, stream>>>(pooled, X, gids, N_NODES, cshift);
    pred_gemm<<<16, 128, 0, stream>>>(pooled, wpack + woff_pred[layer],
                                      (const float*)d_in[PRED_B + layer], score,
                                      din, accumulate);
  };

  // ---- hidden[0] = h ----
  pool_pred(h, IN_DIM, 0, /*accumulate=*/0);

  const float* X = h;
  int din = IN_DIM;
  for (int l = 0; l < 3; l++) {
    const int base = 1 + 22 * l;
    // Y := free buffer (for l==0 that's bufA; afterwards the non-X buffer)
    float* Y   = (X == (const float*)bufA) ? bufB : bufA;
    float* alt = (Y == bufA) ? bufB : bufA;

    // GINConv aggregation: Y = X + sum_{(s,d)} X[s] -> Y[d]
    int cshift = (din == IN_DIM) ? 5 : 6;
    int chunks = din / 4;
    copy_f32v4<<<ceil_div(N_NODES * chunks, 256), 256, 0, stream>>>(Y, X, N_NODES * chunks);
    edge_scatter<<<ceil_div(N_EDGES << cshift, 256), 256, 0, stream>>>(Y, X, esrc, edst,
                                                                       N_EDGES, cshift);

    // 5-layer MLP: (Linear -> BN -> ReLU) x4, then Linear
    float* cur = Y;
    float* nxt = alt;
    for (int i = 0; i < 5; i++) {
      int K = (i == 0) ? din : HID;
      gemm_wmma_f32<<<ceil_div(N_NODES, 64), 256, 0, stream>>>(cur, wpack + woff_gin[l][i],
                                                               nxt, N_NODES, K);
      if (i < 4)
        bn_pass(nxt, (const float*)d_in[base + 10 + i], (const float*)d_in[base + 14 + i]);
      float* t = cur; cur = nxt; nxt = t;
    }
    // ApplyNodeFunc BN -> ReLU, then per-layer BN -> ReLU
    bn_pass(cur, (const float*)d_in[base + 18], (const float*)d_in[base + 19]);
    bn_pass(cur, (const float*)d_in[base + 20], (const float*)d_in[base + 21]);

    X = cur;
    din = HID;
    pool_pred(X, HID, l + 1, /*accumulate=*/1);
  }
}